// SSDLite_86552180949315
// MI455X (gfx1250) — compile-verified
//
#include <hip/hip_runtime.h>
#include <math.h>

// Match XLA: no mul+add -> fma contraction (greedy NMS is numerically chaotic,
// keep op-for-op identical to the reference).
#pragma clang fp contract(off)

#define NUM_CLASSES 21
#define TOP_K       200
#define NPRIORS     3000
#define NPAD        3072          // 12 * NTHREADS, uniform trip count
#define BATCH       256
#define CONF_TH     0.01f
#define NMS_TH      0.45f
#define NTHREADS    256           // 8 x wave32
#define NWAVES      (NTHREADS / 32)
#define ITERS       (NPAD / NTHREADS)   // 12

__global__ __launch_bounds__(NTHREADS)
void ssd_nms_kernel(const float* __restrict__ loc,
                    const float* __restrict__ conf,
                    const float* __restrict__ priors,
                    float* __restrict__ out)
{
    const int bc  = blockIdx.x;              // 0 .. BATCH*NUM_CLASSES-1
    const int b   = bc / NUM_CLASSES;
    const int c   = bc % NUM_CLASSES;
    const int tid = threadIdx.x;

    float* outBase = out + (size_t)bc * (TOP_K * 5);

    // Background class: all zeros.
    if (c == 0) {
        for (int k = tid; k < TOP_K * 5; k += NTHREADS) outBase[k] = 0.0f;
        return;
    }

    // 48KB boxes + 12KB scores (padded) + reduction scratch; WGP has 320KB LDS.
    __shared__ float4 sBox[NPAD];
    __shared__ float  sSc[NPAD];
    __shared__ float  sRedV[NWAVES];
    __shared__ int    sRedI[NWAVES];

    const float4* loc4 = (const float4*)loc;     // [B*P] of (cx,cy,w,h) deltas
    const float4* pri4 = (const float4*)priors;  // [P]   of (cx,cy,w,h)

    // ---- Decode boxes + load masked scores into LDS (global_load_b128) ----
    #pragma unroll
    for (int k = 0; k < ITERS; ++k) {
        const int j = tid + k * NTHREADS;
        if (j < NPRIORS) {
            float4 l = loc4[b * NPRIORS + j];
            float4 p = pri4[j];
            float cx = p.x + (l.x * 0.1f) * p.z;   // priors_cxy + loc*VAR0*priors_wh
            float cy = p.y + (l.y * 0.1f) * p.w;
            float w  = p.z * expf(l.z * 0.2f);     // priors_wh * exp(loc*VAR1)
            float h  = p.w * expf(l.w * 0.2f);
            float4 bb;
            bb.x = cx - w * 0.5f;
            bb.y = cy - h * 0.5f;
            bb.z = cx + w * 0.5f;
            bb.w = cy + h * 0.5f;
            sBox[j] = bb;
            float s = conf[((size_t)b * NPRIORS + j) * NUM_CLASSES + c];
            sSc[j]  = (s > CONF_TH) ? s : -INFINITY;
        } else {
            // Padding: can never win argmax; suppression only writes -inf.
            sBox[j] = make_float4(0.0f, 0.0f, 0.0f, 0.0f);
            sSc[j]  = -INFINITY;
        }
    }
    __syncthreads();

    const int wid  = tid >> 5;
    const int lane = tid & 31;

    for (int t = 0; t < TOP_K; ++t) {
        // ---------- block-wide argmax (first-occurrence tie-break) ----------
        float v   = -INFINITY;
        int   idx = 0x7fffffff;
        #pragma unroll
        for (int k = 0; k < ITERS; ++k) {
            const int j = tid + k * NTHREADS;
            float s = sSc[j];
            if (s > v || (s == v && j < idx)) { v = s; idx = j; }
        }
        // wave32 xor-shuffle reduction
        #pragma unroll
        for (int off = 16; off > 0; off >>= 1) {
            float ov = __shfl_xor(v,   off, 32);
            int   oi = __shfl_xor(idx, off, 32);
            if (ov > v || (ov == v && oi < idx)) { v = ov; idx = oi; }
        }
        if (lane == 0) { sRedV[wid] = v; sRedI[wid] = idx; }
        __syncthreads();

        // Every thread reduces the 8 per-wave results (uniform, no 2nd barrier).
        float bestV = sRedV[0];
        int   bestI = sRedI[0];
        #pragma unroll
        for (int wq = 1; wq < NWAVES; ++wq) {
            float ov = sRedV[wq]; int oi = sRedI[wq];
            if (ov > bestV || (ov == bestV && oi < bestI)) { bestV = ov; bestI = oi; }
        }

        // Scores are either > CONF_TH or -inf; once max is -inf it stays -inf.
        if (!(bestV > CONF_TH)) {
            const int n = (TOP_K - t) * 5;
            float* o = outBase + t * 5;
            for (int k = tid; k < n; k += NTHREADS) o[k] = 0.0f;
            return;  // uniform branch: all threads exit together
        }

        const float4 B = sBox[bestI];                   // LDS broadcast read
        const float area1 = (B.z - B.x) * (B.w - B.y);  // same as reference

        if (tid == 0) {
            float* o = outBase + t * 5;
            o[0] = bestV; o[1] = B.x; o[2] = B.y; o[3] = B.z; o[4] = B.w;
            sSc[bestI] = -INFINITY;  // self-IoU==1 also suppresses; benign dup
        }

        // ---------- IoU suppression against all priors (uniform, unrolled) ----
        #pragma unroll
        for (int k = 0; k < ITERS; ++k) {
            const int j = tid + k * NTHREADS;
            float4 bb  = sBox[j];                        // ds_load_b128
            float ltx  = fmaxf(B.x, bb.x);
            float lty  = fmaxf(B.y, bb.y);
            float rbx  = fminf(B.z, bb.z);
            float rby  = fminf(B.w, bb.w);
            float w    = fmaxf(rbx - ltx, 0.0f);
            float h    = fmaxf(rby - lty, 0.0f);
            float inter = w * h;
            float area2 = (bb.z - bb.x) * (bb.w - bb.y);
            float iou   = inter / ((area1 + area2) - inter);  // IEEE div, ref order
            if (iou > NMS_TH) sSc[j] = -INFINITY;
        }
        __syncthreads();
    }
}

extern "C" void kernel_launch(void* const* d_in, const int* in_sizes, int n_in,
                              void* d_out, int out_size, void* d_ws, size_t ws_size,
                              hipStream_t stream) {
    (void)in_sizes; (void)n_in; (void)d_ws; (void)ws_size; (void)out_size;
    const float* loc    = (const float*)d_in[0];  // [B, P, 4]
    const float* conf   = (const float*)d_in[1];  // [B, P, 21]
    const float* priors = (const float*)d_in[2];  // [P, 4]
    float* out = (float*)d_out;                   // [B, 21, TOP_K, 5]

    dim3 grid(BATCH * NUM_CLASSES);
    dim3 block(NTHREADS);
    ssd_nms_kernel<<<grid, block, 0, stream>>>(loc, conf, priors, out);
}